// CIF_36369783063100
// MI455X (gfx1250) — compile-verified
//
#include <hip/hip_runtime.h>
#include <math.h>

// Problem constants (match reference)
#define B_  32
#define T_  2000
#define H_  512
#define L_  256
#define TH_ 0.95f

typedef float v2f __attribute__((ext_vector_type(2)));
typedef float v8f __attribute__((ext_vector_type(8)));

// Packed per-frame routing record: frame t sends weight wa to token ta and
// (on a fire) weight wb to token tb. 16 bytes -> single b128 load/store.
struct alignas(16) Ftok {
  float wa, wb;
  int   ta, tb;
};

// --- CDNA5 async load-to-LDS path (probe via __has_builtin) ----------------
#if defined(__has_builtin)
#if __has_builtin(__builtin_amdgcn_global_load_async_to_lds_b128)
#define CIF_ASYNC_LDS 1
#endif
#endif
#ifndef CIF_ASYNC_LDS
#define CIF_ASYNC_LDS 0
#endif

#if CIF_ASYNC_LDS
// Builtin prototype (from hipcc diagnostic): params are pointers to a
// 16-byte int vector in AS(1) (global) / AS(3) (LDS).
typedef int v4i_ __attribute__((vector_size(16)));
typedef __attribute__((address_space(1))) v4i_ as1_v4i;
typedef __attribute__((address_space(3))) v4i_ as3_v4i;
#if __has_builtin(__builtin_amdgcn_s_wait_asynccnt)
#define CIF_WAIT_ASYNC(n) __builtin_amdgcn_s_wait_asynccnt(n)
#else
#define CIF_WAIT_ASYNC(n) asm volatile("s_wait_asynccnt %0" ::"i"(n) : "memory")
#endif
#endif

// ---------------------------------------------------------------------------
// Phase 1: one wave per batch. Rescale alphas, wave-parallel prefix-sum,
// derive per-frame routing records + fire-frame table + valid token count.
// ---------------------------------------------------------------------------
__global__ void __launch_bounds__(32) cif_scan_kernel(
    const float* __restrict__ alphas, const int* __restrict__ tlen,
    Ftok* __restrict__ ftok, int* __restrict__ fireframe,
    int* __restrict__ nfinArr)
{
  const int b = blockIdx.x;
  const int lane = (int)threadIdx.x;
  const float* arow = alphas + b * T_;

  float sum = 0.f;
  for (int t = lane; t < T_; t += 32) sum += arow[t];
#pragma unroll
  for (int d = 16; d > 0; d >>= 1) sum += __shfl_down(sum, d, 32);
  sum = __shfl(sum, 0, 32);
  const float scale = (float)tlen[b] / sum;

  float carry = 0.f;
  int Flast = 0;
  for (int t0 = 0; t0 < T_; t0 += 32) {
    const int t = t0 + lane;
    const float a = (t < T_) ? arow[t] * scale : 0.f;

    float v = a;                          // inclusive wave prefix sum
#pragma unroll
    for (int d = 1; d < 32; d <<= 1) {
      float u = __shfl_up(v, d, 32);
      if (lane >= d) v += u;
    }
    const float pre  = carry + v;         // prefix through t
    const float prev = pre - a;           // prefix through t-1

    // Closed-form fire count (valid: every scaled alpha << 1).
    const int F  = (pre  >= TH_) ? ((int)floorf(pre  - TH_) + 1) : 0;
    const int Fp = (prev >= TH_) ? ((int)floorf(prev - TH_) + 1) : 0;
    const bool fire = F > Fp;

    const float dist = 1.0f - (prev - (float)Fp);   // dist_completion
    const float cur  = fire ? dist : a;
    const float rem  = a - cur;

    if (t < T_) {
      Ftok f;
      f.wa = cur;
      f.wb = fire ? rem : 0.f;
      f.ta = Fp;
      f.tb = fire ? (Fp + 1) : -1;
      ftok[b * T_ + t] = f;
      if (fire && Fp < L_) fireframe[b * L_ + Fp] = t;
    }
    carry = __shfl(pre, 31, 32);
    Flast = __shfl(F,  31, 32);
  }
  if (lane == 0) nfinArr[b] = (Flast < L_) ? Flast : L_;
}

__device__ __forceinline__ float cif_wrow(const Ftok& f, int row) {
  return (f.ta == row ? f.wa : 0.f) + (f.tb == row ? f.wb : 0.f);
}

// Stage 4 frames (8 KB, contiguous) into an LDS buffer. Frames past T-1 are
// clamped (their A weights are exactly zero), so every thread always issues
// exactly 2 x b128 -> per-wave ASYNCcnt bookkeeping stays deterministic.
#if CIF_ASYNC_LDS
__device__ __forceinline__ void cif_stage(const float* __restrict__ hb,
                                          int k0, float* lds, int tid)
{
#pragma unroll
  for (int i = 0; i < 2; ++i) {
    const int f = tid + (i << 8);                  // float4 slot 0..511
    int t = k0 + (f >> 7);                         // 128 float4 per frame
    t = (t < T_) ? t : (T_ - 1);
    const float* g = hb + (size_t)t * H_ + ((f & 127) << 2);
    float* l = lds + (f << 2);
    __builtin_amdgcn_global_load_async_to_lds_b128(
        (as1_v4i*)(uintptr_t)g,
        (as3_v4i*)(uint32_t)(uintptr_t)l, 0, 0);
  }
}
#else
__device__ __forceinline__ void cif_stage(const float* __restrict__ hb,
                                          int k0, float* lds, int tid)
{
#pragma unroll
  for (int i = 0; i < 2; ++i) {
    const int f = tid + (i << 8);
    int t = k0 + (f >> 7);
    t = (t < T_) ? t : (T_ - 1);
    const float4 v = *((const float4*)(hb + (size_t)t * H_) + (f & 127));
    *((float4*)lds + f) = v;
  }
}
#endif

// ---------------------------------------------------------------------------
// Phase 2: banded sparse GEMM out[L,H] = W[L,T] * hidden[T,H] per batch as
// fp32 WMMA 16x16x4. One WG per (batch, 16-token tile); 8 waves x 64 ch.
// Double-buffered LDS; hidden streamed essentially once, async when possible.
// ---------------------------------------------------------------------------
__global__ void __launch_bounds__(256) cif_wmma_kernel(
    const float* __restrict__ hidden,
    const Ftok* __restrict__ ftok,
    const int* __restrict__ fireframe, const int* __restrict__ nfinArr,
    float* __restrict__ out)
{
  const int b    = blockIdx.y;
  const int l0   = blockIdx.x << 4;
  const int tid  = (int)threadIdx.x;
  const int wave = tid >> 5;
  const int lane = tid & 31;

  __shared__ alignas(16) float sh[2][4 * H_];      // 2 x 8 KB

  const int nfin = nfinArr[b];
  int s = 0, e = -1;                               // inclusive frame range
  if (l0 < nfin) {
    s = (l0 == 0) ? 0 : fireframe[b * L_ + (l0 - 1)];
    int lastv = l0 + 15;
    if (lastv > nfin - 1) lastv = nfin - 1;
    e = fireframe[b * L_ + lastv];
  }

  v8f acc[4];
#pragma unroll
  for (int c = 0; c < 4; ++c)
#pragma unroll
    for (int r = 0; r < 8; ++r) acc[c][r] = 0.f;

  // WMMA fragment coords (A 16x4 f32: lane<16 -> K0/K1, lane>=16 -> K2/K3)
  const int  m        = lane & 15;
  const int  khalf    = (lane >> 4) << 1;
  const int  row      = l0 + m;
  const bool rowvalid = row < nfin;
  const int  base     = b * T_;
  const float* hb     = hidden + (size_t)b * T_ * H_;

  if (s <= e) cif_stage(hb, s, sh[0], tid);        // prefetch first chunk

  int ibuf = 0;
  for (int k0 = s; k0 <= e; k0 += 4, ibuf ^= 1) {
    const bool more = (k0 + 4) <= e;
    if (more) cif_stage(hb, k0 + 4, sh[ibuf ^ 1], tid);   // prefetch next

    // Routing records for this lane's two K columns (overlaps staging).
    const int t0 = k0 + khalf, t1 = t0 + 1;
    const Ftok f0 = ftok[base + ((t0 < T_) ? t0 : (T_ - 1))];
    const Ftok f1 = ftok[base + ((t1 < T_) ? t1 : (T_ - 1))];

#if CIF_ASYNC_LDS
    if (more) { CIF_WAIT_ASYNC(2); } else { CIF_WAIT_ASYNC(0); }
#endif
    __syncthreads();                               // chunk k0 visible to all

    v2f Af;
    Af.x = (rowvalid && t0 < T_) ? cif_wrow(f0, row) : 0.f;
    Af.y = (rowvalid && t1 < T_) ? cif_wrow(f1, row) : 0.f;

    const float* cur = sh[ibuf];
#pragma unroll
    for (int c = 0; c < 4; ++c) {
      const int ch0 = (wave << 6) + (c << 4);
      v2f Bf;
      Bf.x = cur[(khalf    ) * H_ + ch0 + m];
      Bf.y = cur[(khalf + 1) * H_ + ch0 + m];
      acc[c] = __builtin_amdgcn_wmma_f32_16x16x4_f32(
          false, Af, false, Bf, (short)0, acc[c], false, false);
    }
    __syncthreads();                               // done reading sh[ibuf]
  }

  // D layout: VGPR r -> M=r (lanes 0-15) / M=r+8 (lanes 16-31), N = lane&15.
  const int rbase = (lane >> 4) << 3;
#pragma unroll
  for (int c = 0; c < 4; ++c) {
    const int ch0 = (wave << 6) + (c << 4);
#pragma unroll
    for (int r = 0; r < 8; ++r) {
      const int l = l0 + rbase + r;
      __builtin_nontemporal_store(acc[c][r],
          &out[((size_t)b * L_ + l) * H_ + ch0 + m]);
    }
  }
}

// ---------------------------------------------------------------------------
extern "C" void kernel_launch(void* const* d_in, const int* in_sizes, int n_in,
                              void* d_out, int out_size, void* d_ws, size_t ws_size,
                              hipStream_t stream)
{
  (void)in_sizes; (void)n_in; (void)out_size; (void)ws_size;
  const float* hidden = (const float*)d_in[0];   // [B,T,H] f32
  const float* alphas = (const float*)d_in[1];   // [B,T]   f32
  const int*   tlen   = (const int*)  d_in[2];   // [B]     i32
  float* out = (float*)d_out;                    // [B,L,H] f32

  // Workspace layout (~1.06 MB)
  char* ws = (char*)d_ws;
  Ftok* ftok       = (Ftok*)ws;                  // B*T * 16B
  int*  fireframe  = (int*)(ftok + B_ * T_);     // B*L
  int*  nfinArr    = fireframe + B_ * L_;        // B

  cif_scan_kernel<<<B_, 32, 0, stream>>>(alphas, tlen, ftok, fireframe, nfinArr);

  dim3 grid(L_ / 16, B_);   // 16 token-tiles x 32 batches
  cif_wmma_kernel<<<grid, 256, 0, stream>>>(hidden, ftok, fireframe, nfinArr, out);
}